// SolarMoEReference_10840497455878
// MI455X (gfx1250) — compile-verified
//
#include <hip/hip_runtime.h>
#include <math.h>

#define H_DIM 2048
#define I_DIM 768
#define SI_DIM 2048
#define E_NUM 128
#define TOPK 8
#define ROUTED_SCALE 2.5f

#define UP_SPLIT 4
#define UP_CHUNK (H_DIM / UP_SPLIT)      // 512
#define N_UP_TILES 512                   // 8*48 routed + 128 shared
#define NSEG 12                          // 8 expert segs + 4 shared chunks

typedef float v2f __attribute__((ext_vector_type(2)));
typedef float v4f __attribute__((ext_vector_type(4)));
typedef float v8f __attribute__((ext_vector_type(8)));

// ---- workspace layout (bytes) ----
static constexpr size_t WS_IDX   = 0;                        // int[8]
static constexpr size_t WS_WK    = 64;                       // float[8]
static constexpr size_t WS_G1P   = 256;                      // float[512*4*16]
static constexpr size_t G_P_ELEM = (size_t)N_UP_TILES * UP_SPLIT * 16;
static constexpr size_t WS_G3P   = WS_G1P + G_P_ELEM * 4;
static constexpr size_t WS_T     = WS_G3P + G_P_ELEM * 4;    // float[8192]
static constexpr size_t WS_DP    = WS_T + 8192 * 4;          // float[1536*16]

// D = A(16x4) x B(4x16) + C accumulate; k-permuted 8-wide step:
// half0 lanes carry logical k = kb..kb+3, half1 lanes carry kb+4..kb+7.
// wmma #0 uses .xy of each float4, wmma #1 uses .zw — each logical k
// appears exactly once, and A/B share the mapping, so the dot is exact.
__device__ __forceinline__ void wmma_step8(v8f& acc, v4f a, v4f b) {
  v2f a0 = __builtin_shufflevector(a, a, 0, 1);
  v2f a1 = __builtin_shufflevector(a, a, 2, 3);
  v2f b0 = __builtin_shufflevector(b, b, 0, 1);
  v2f b1 = __builtin_shufflevector(b, b, 2, 3);
  acc = __builtin_amdgcn_wmma_f32_16x16x4_f32(false, a0, false, b0, (short)0, acc, false, false);
  acc = __builtin_amdgcn_wmma_f32_16x16x4_f32(false, a1, false, b1, (short)0, acc, false, false);
}

// ------------------------------------------------------------------
// 1) gating: logits -> sigmoid -> biased top-8 -> normalized weights
// ------------------------------------------------------------------
__global__ void gate_topk_kernel(const float* __restrict__ x,
                                 const float* __restrict__ gw,
                                 const float* __restrict__ bias,
                                 int* __restrict__ idx,
                                 float* __restrict__ wk) {
  __shared__ float sc[E_NUM];
  __shared__ float bi[E_NUM];
  int e = threadIdx.x;
  const float* row = gw + (size_t)e * H_DIM;
  float acc = 0.f;
#pragma unroll 8
  for (int h = 0; h < H_DIM; ++h) acc = fmaf(row[h], x[h], acc);
  float s = 1.f / (1.f + expf(-acc));
  sc[e] = s;
  bi[e] = s + bias[e];
  __syncthreads();
  if (e == 0) {
    int sel[TOPK];
    float sum = 0.f;
    for (int k = 0; k < TOPK; ++k) {
      int best = 0; float bv = -1e30f;
      for (int j = 0; j < E_NUM; ++j)
        if (bi[j] > bv) { bv = bi[j]; best = j; }   // ties -> lowest index
      sel[k] = best;
      bi[best] = -1e30f;
      sum += sc[best];
    }
    float inv = ROUTED_SCALE / (sum + 1e-20f);
    for (int k = 0; k < TOPK; ++k) { idx[k] = sel[k]; wk[k] = sc[sel[k]] * inv; }
  }
}

// ------------------------------------------------------------------
// 2) up/gate matvec partials: one b128/lane per matrix per 8 k-values
// ------------------------------------------------------------------
__global__ void up_partial_kernel(const float* __restrict__ x,
                                  const float* __restrict__ ew1,
                                  const float* __restrict__ ew3,
                                  const float* __restrict__ sw1,
                                  const float* __restrict__ sw3,
                                  const int* __restrict__ idx,
                                  float* __restrict__ g1p,
                                  float* __restrict__ g3p) {
  __shared__ float xl[H_DIM];
  for (int i = threadIdx.x; i < H_DIM; i += blockDim.x) xl[i] = x[i];
  __syncthreads();

  const int wave = blockIdx.x * (blockDim.x >> 5) + (threadIdx.x >> 5);
  const int lane = threadIdx.x & 31;
  const int tile = wave / UP_SPLIT;
  const int chunk = wave % UP_SPLIT;
  const int half = lane >> 4;
  const int m = lane & 15;

  const float *wg, *wu; int row0;
  if (tile < 8 * (I_DIM / 16)) {   // routed tiles
    int k = tile / (I_DIM / 16);
    int e = idx[k];
    row0 = (tile % (I_DIM / 16)) * 16;
    size_t base = (size_t)e * I_DIM * H_DIM;
    wg = ew1 + base; wu = ew3 + base;
  } else {                         // shared expert tiles
    row0 = (tile - 8 * (I_DIM / 16)) * 16;
    wg = sw1; wu = sw3;
  }
  const float* pg = wg + (size_t)(row0 + m) * H_DIM + 4 * half;
  const float* pu = wu + (size_t)(row0 + m) * H_DIM + 4 * half;
  const float* px = xl + 4 * half;

  v8f cg = {}; v8f cu = {};
  const int kb0 = chunk * UP_CHUNK;
#pragma unroll 2
  for (int kb = kb0; kb < kb0 + UP_CHUNK; kb += 8) {
    v4f a1 = *(const v4f*)(pg + kb);
    v4f a3 = *(const v4f*)(pu + kb);
    v4f b  = *(const v4f*)(px + kb);
    wmma_step8(cg, a1, b);
    wmma_step8(cu, a3, b);
  }
  // D columns identical; lane 0 holds rows 0..7, lane 16 holds rows 8..15
  float* o1 = g1p + ((size_t)tile * UP_SPLIT + chunk) * 16;
  float* o3 = g3p + ((size_t)tile * UP_SPLIT + chunk) * 16;
  if (lane == 0 || lane == 16) {
    int rb = half * 8;
#pragma unroll
    for (int v = 0; v < 8; ++v) { o1[rb + v] = cg[v]; o3[rb + v] = cu[v]; }
  }
}

// ------------------------------------------------------------------
// 3) combine chunk partials, apply silu(g)*u, fold routed weight
// ------------------------------------------------------------------
__global__ void act_kernel(const float* __restrict__ wk,
                           const float* __restrict__ g1p,
                           const float* __restrict__ g3p,
                           float* __restrict__ t) {
  int i = blockIdx.x * blockDim.x + threadIdx.x;
  if (i >= 8192) return;
  int tile, r; float scale;
  if (i < 8 * I_DIM) {
    int k = i / I_DIM, ii = i % I_DIM;
    tile = k * (I_DIM / 16) + ii / 16; r = ii & 15;
    scale = wk[k];
  } else {
    int j = i - 8 * I_DIM;
    tile = 8 * (I_DIM / 16) + j / 16; r = j & 15;
    scale = 1.f;
  }
  float g = 0.f, u = 0.f;
#pragma unroll
  for (int c = 0; c < UP_SPLIT; ++c) {
    g += g1p[((size_t)tile * UP_SPLIT + c) * 16 + r];
    u += g3p[((size_t)tile * UP_SPLIT + c) * 16 + r];
  }
  float silu = g / (1.f + expf(-g));
  t[i] = scale * silu * u;
}

// ------------------------------------------------------------------
// 4) down-projection partials (WMMA matvec), t staged in LDS
// ------------------------------------------------------------------
__global__ void down_partial_kernel(const float* __restrict__ ew2,
                                    const float* __restrict__ sw2,
                                    const int* __restrict__ idx,
                                    const float* __restrict__ t,
                                    float* __restrict__ dp) {
  __shared__ float tl[8192];
  for (int i = threadIdx.x; i < 8192; i += blockDim.x) tl[i] = t[i];
  __syncthreads();

  const int wave = blockIdx.x * (blockDim.x >> 5) + (threadIdx.x >> 5);
  const int lane = threadIdx.x & 31;
  const int rt = wave / NSEG;
  const int s = wave % NSEG;
  const int half = lane >> 4;
  const int m = lane & 15;
  const int row0 = rt * 16;

  const float* w; const float* bsrc; size_t ld; int kb0, kcnt;
  if (s < 8) {                       // one routed expert, full I
    int e = idx[s];
    w = ew2 + (size_t)e * H_DIM * I_DIM;
    ld = I_DIM;
    bsrc = tl + s * I_DIM;
    kb0 = 0; kcnt = I_DIM;
  } else {                           // shared expert, 512-wide chunk
    w = sw2; ld = SI_DIM;
    bsrc = tl + 8 * I_DIM;
    kb0 = (s - 8) * 512; kcnt = 512;
  }
  const float* pw = w + (size_t)(row0 + m) * ld + 4 * half;
  const float* pb = bsrc + 4 * half;

  v8f c = {};
#pragma unroll 4
  for (int kb = kb0; kb < kb0 + kcnt; kb += 8) {
    v4f a = *(const v4f*)(pw + kb);
    v4f b = *(const v4f*)(pb + kb);
    wmma_step8(c, a, b);
  }
  float* o = dp + (size_t)wave * 16;
  if (lane == 0 || lane == 16) {
    int rb = half * 8;
#pragma unroll
    for (int v = 0; v < 8; ++v) o[rb + v] = c[v];
  }
}

// ------------------------------------------------------------------
// 5) fixed-order final reduction over the 12 segments
// ------------------------------------------------------------------
__global__ void combine_kernel(const float* __restrict__ dp,
                               float* __restrict__ out) {
  int h = blockIdx.x * blockDim.x + threadIdx.x;
  if (h >= H_DIM) return;
  int rt = h / 16, r = h & 15;
  float acc = 0.f;
#pragma unroll
  for (int s = 0; s < NSEG; ++s)
    acc += dp[((size_t)(rt * NSEG + s)) * 16 + r];
  out[h] = acc;
}

extern "C" void kernel_launch(void* const* d_in, const int* in_sizes, int n_in,
                              void* d_out, int out_size, void* d_ws, size_t ws_size,
                              hipStream_t stream) {
  const float* x    = (const float*)d_in[0];
  const float* gw   = (const float*)d_in[1];
  const float* bias = (const float*)d_in[2];
  const float* ew1  = (const float*)d_in[3];
  const float* ew2  = (const float*)d_in[4];
  const float* ew3  = (const float*)d_in[5];
  const float* sw1  = (const float*)d_in[6];
  const float* sw2  = (const float*)d_in[7];
  const float* sw3  = (const float*)d_in[8];
  float* out = (float*)d_out;

  char* ws = (char*)d_ws;
  int*   idx = (int*)(ws + WS_IDX);
  float* wk  = (float*)(ws + WS_WK);
  float* g1p = (float*)(ws + WS_G1P);
  float* g3p = (float*)(ws + WS_G3P);
  float* t   = (float*)(ws + WS_T);
  float* dp  = (float*)(ws + WS_DP);

  gate_topk_kernel<<<1, 128, 0, stream>>>(x, gw, bias, idx, wk);
  // 2048 waves = 256 blocks * 8 waves
  up_partial_kernel<<<256, 256, 0, stream>>>(x, ew1, ew3, sw1, sw3, idx, g1p, g3p);
  act_kernel<<<32, 256, 0, stream>>>(wk, g1p, g3p, t);
  // 1536 waves = 192 blocks * 8 waves
  down_partial_kernel<<<192, 256, 0, stream>>>(ew2, sw2, idx, t, dp);
  combine_kernel<<<8, 256, 0, stream>>>(dp, out);
}